// GNN_17746804867649
// MI455X (gfx1250) — compile-verified
//
#include <hip/hip_runtime.h>
#include <math.h>

typedef __attribute__((ext_vector_type(2))) float v2f;
typedef __attribute__((ext_vector_type(8))) float v8f;

#define NN   50000
#define NE   800000
#define CH   128
#define NOUT 40
#define LN_EPS 1e-5f
#define KPAD (CH + 4)   // LDS k-stride in floats: 8B aligned, avoids 64-bank wrap

// ---------------------------------------------------------------------------
// degree: deg starts at 2 (self loop added twice in reference), then +1 per
// in-edge; dinv = rsqrt(deg) (deg >= 2 so the >0 / max(.,1) guards are moot).
// ---------------------------------------------------------------------------
__global__ void k_deg_init(float* __restrict__ deg) {
    int i = blockIdx.x * blockDim.x + threadIdx.x;
    if (i < NN) deg[i] = 2.0f;
}

__global__ void k_deg_accum(const int* __restrict__ dst, float* __restrict__ deg) {
    int e = blockIdx.x * blockDim.x + threadIdx.x;
    if (e < NE) atomicAdd(&deg[dst[e]], 1.0f);
}

__global__ void k_dinv(float* __restrict__ deg) {
    int i = blockIdx.x * blockDim.x + threadIdx.x;
    if (i < NN) deg[i] = rsqrtf(deg[i]);
}

__global__ void k_zero(float* __restrict__ p, long long n) {
    long long i = (long long)blockIdx.x * blockDim.x + threadIdx.x;
    if (i < n) p[i] = 0.0f;
}

// ---------------------------------------------------------------------------
// fp32 WMMA GEMM: C[M x nvalid] = A[M x 128] * W[128 x nvalid], row-major.
// Block = 32*nColTiles threads = one 16-row strip, one wave per 16x16 tile.
// W is staged in LDS transposed (Wt[col][k], stride KPAD) and zero-padded to
// 16*nColTiles columns, so the inner loop is branch-free:
//   global_load_b64 (A frag) + ds_load_b64 (B frag) + v_wmma_f32_16x16x4_f32.
// A frag: lane l<16 -> A[rt*16+l][k+0..1], lane l+16 -> K = k+2..3.
// B frag: lane l<16 -> W[k+0..1][ct*16+l], lane l+16 -> K = k+2..3.
// D: VGPR r -> M = rt*16 + r + 8*(lane>=16), N = ct*16 + (lane&15).
// ---------------------------------------------------------------------------
__global__ void k_wmma_gemm(const float* __restrict__ A, const float* __restrict__ W,
                            float* __restrict__ C, int ldw, int nvalid, int ldc) {
    extern __shared__ float lds[];          // [16*nColTiles][KPAD], transposed W
    const int ncols16 = blockDim.x >> 1;    // 16 * nColTiles  (blockDim = 32*nCT)
    const int rowTile = blockIdx.x;

    // cooperative transpose-fill: consecutive threads read consecutive cols of
    // one W row (coalesced); cols >= nvalid are zero-padded.
    for (int idx = threadIdx.x; idx < CH * ncols16; idx += blockDim.x) {
        int k   = idx / ncols16;
        int col = idx - k * ncols16;
        lds[(size_t)col * KPAD + k] = (col < nvalid) ? W[(size_t)k * ldw + col] : 0.0f;
    }
    __syncthreads();

    const int colTile = threadIdx.x >> 5;   // wave id
    const int lane = threadIdx.x & 31;
    const int half = lane >> 4;             // 0: K pair {0,1}, 1: K pair {2,3}
    const int l16  = lane & 15;

    const int arow = rowTile * 16 + l16;    // A row this lane streams
    const int lcol = colTile * 16 + l16;    // local (padded) B/C column

    const float* ap = A + (size_t)arow * CH + 2 * half;
    const float* wt = lds + (size_t)lcol * KPAD + 2 * half;

    v8f acc = {};
#pragma unroll
    for (int k = 0; k < CH; k += 4) {
        v2f a = *reinterpret_cast<const v2f*>(ap + k);   // A[arow][k+2h .. +1]
        v2f b = *reinterpret_cast<const v2f*>(wt + k);   // W[k+2h .. +1][col]
        acc = __builtin_amdgcn_wmma_f32_16x16x4_f32(
            /*neg_a=*/false, a, /*neg_b=*/false, b,
            /*c_mod=*/(short)0, acc, /*reuse_a=*/false, /*reuse_b=*/false);
    }

    if (lcol < nvalid) {
        float* cp = C + (size_t)(rowTile * 16 + 8 * half) * ldc + lcol;
#pragma unroll
        for (int r = 0; r < 8; ++r)
            cp[(size_t)r * ldc] = acc[r];
    }
}

// ---------------------------------------------------------------------------
// Edge scatter: one wave per edge, 4 channels per lane (float4 gather +
// 4 global atomic f32 adds). Bandwidth-bound; dominates runtime.
// ---------------------------------------------------------------------------
__global__ void k_scatter(const int* __restrict__ src, const int* __restrict__ dst,
                          const float* __restrict__ dinv, const float* __restrict__ hm,
                          float* __restrict__ acc) {
    int e = blockIdx.x * (blockDim.x >> 5) + (threadIdx.x >> 5);
    if (e >= NE) return;
    int lane = threadIdx.x & 31;
    int s = src[e], d = dst[e];
    float w = dinv[s] * dinv[d];
    float4 hv = reinterpret_cast<const float4*>(hm + (size_t)s * CH)[lane];
    float* ap = acc + (size_t)d * CH + lane * 4;
    atomicAdd(ap + 0, w * hv.x);
    atomicAdd(ap + 1, w * hv.y);
    atomicAdd(ap + 2, w * hv.z);
    atomicAdd(ap + 3, w * hv.w);
}

// ---------------------------------------------------------------------------
// Epilogue: acc += 2*dinv^2*hm (double self loop) + bias; ReLU; LayerNorm;
// optional residual. One wave per row, 4 channels/lane, shuffle reductions.
// Writes in-place into acc.
// ---------------------------------------------------------------------------
__global__ void k_epilogue(float* __restrict__ acc, const float* __restrict__ hm,
                           const float* __restrict__ res, const float* __restrict__ dinv,
                           const float* __restrict__ b, const float* __restrict__ g,
                           const float* __restrict__ beta, int add_res) {
    int row = blockIdx.x * (blockDim.x >> 5) + (threadIdx.x >> 5);
    if (row >= NN) return;
    int lane = threadIdx.x & 31;
    float di = dinv[row];
    float selfw = 2.0f * di * di;

    float v[4];
    float s1 = 0.0f, s2 = 0.0f;
#pragma unroll
    for (int j = 0; j < 4; ++j) {
        int c = lane + 32 * j;
        float t = acc[(size_t)row * CH + c] + selfw * hm[(size_t)row * CH + c] + b[c];
        t = fmaxf(t, 0.0f);
        v[j] = t;
        s1 += t;
        s2 += t * t;
    }
#pragma unroll
    for (int off = 16; off >= 1; off >>= 1) {
        s1 += __shfl_xor(s1, off, 32);
        s2 += __shfl_xor(s2, off, 32);
    }
    float mu = s1 * (1.0f / CH);
    float var = s2 * (1.0f / CH) - mu * mu;
    float rstd = rsqrtf(var + LN_EPS);
#pragma unroll
    for (int j = 0; j < 4; ++j) {
        int c = lane + 32 * j;
        float o = (v[j] - mu) * rstd * g[c] + beta[c];
        if (add_res) o += res[(size_t)row * CH + c];
        acc[(size_t)row * CH + c] = o;
    }
}

// ---------------------------------------------------------------------------
// Fused bias + log-softmax over 40 logits: one wave per row, 2 logits/lane.
// ---------------------------------------------------------------------------
__global__ void k_logsoftmax(const float* __restrict__ logits, const float* __restrict__ b_out,
                             float* __restrict__ out) {
    int row = blockIdx.x * (blockDim.x >> 5) + (threadIdx.x >> 5);
    if (row >= NN) return;
    int lane = threadIdx.x & 31;
    int c1 = lane + 32;
    bool has1 = c1 < NOUT;
    float x0 = logits[(size_t)row * NOUT + lane] + b_out[lane];
    float x1 = has1 ? logits[(size_t)row * NOUT + c1] + b_out[c1] : -INFINITY;
    float m = fmaxf(x0, x1);
#pragma unroll
    for (int off = 16; off >= 1; off >>= 1)
        m = fmaxf(m, __shfl_xor(m, off, 32));
    float s = __expf(x0 - m) + (has1 ? __expf(x1 - m) : 0.0f);
#pragma unroll
    for (int off = 16; off >= 1; off >>= 1)
        s += __shfl_xor(s, off, 32);
    float lse = m + __logf(s);
    out[(size_t)row * NOUT + lane] = x0 - lse;
    if (has1) out[(size_t)row * NOUT + c1] = x1 - lse;
}

// ---------------------------------------------------------------------------
extern "C" void kernel_launch(void* const* d_in, const int* in_sizes, int n_in,
                              void* d_out, int out_size, void* d_ws, size_t ws_size,
                              hipStream_t stream) {
    const float* x     = (const float*)d_in[0];
    const int*   ei    = (const int*)d_in[1];
    const int*   src   = ei;            // edge_index[0]
    const int*   dst   = ei + NE;       // edge_index[1]
    const float* W0    = (const float*)d_in[2];
    const float* b0    = (const float*)d_in[3];
    const float* g0    = (const float*)d_in[4];
    const float* beta0 = (const float*)d_in[5];
    const float* W1    = (const float*)d_in[6];
    const float* b1    = (const float*)d_in[7];
    const float* g1    = (const float*)d_in[8];
    const float* beta1 = (const float*)d_in[9];
    const float* W2    = (const float*)d_in[10];
    const float* b2    = (const float*)d_in[11];
    const float* g2    = (const float*)d_in[12];
    const float* beta2 = (const float*)d_in[13];
    const float* W_out = (const float*)d_in[14];
    const float* b_out = (const float*)d_in[15];
    float* out = (float*)d_out;

    // workspace: dinv (N) + 3 rotating N x 128 buffers (~77 MB total)
    char* ws = (char*)d_ws;
    float* dinv = (float*)ws;
    size_t off = ((size_t)NN * sizeof(float) + 255) & ~(size_t)255;
    size_t bufBytes = (size_t)NN * CH * sizeof(float);
    float* bufA = (float*)(ws + off);
    float* bufB = (float*)(ws + off + bufBytes);
    float* bufC = (float*)(ws + off + 2 * bufBytes);

    // ---- degrees ----
    k_deg_init<<<(NN + 255) / 256, 256, 0, stream>>>(dinv);
    k_deg_accum<<<(NE + 255) / 256, 256, 0, stream>>>(dst, dinv);
    k_dinv<<<(NN + 255) / 256, 256, 0, stream>>>(dinv);

    const int nRowTiles = NN / 16;                 // 3125 exact
    const long long nElem = (long long)NN * CH;
    const int zeroBlks  = (int)((nElem + 255) / 256);
    const int scatBlks  = (NE + 7) / 8;            // 8 edges / 256-thread block
    const int rowBlks   = (NN + 3) / 4;            // 4 rows / 128-thread block

    const int ldsH = CH * KPAD * sizeof(float);    // 128 cols staged (66 KB)
    const int ldsO = 48 * KPAD * sizeof(float);    // 48 cols staged (25 KB)

    // ---- layer 0: h_in = x, hm = bufA, acc/out = bufB ----
    k_wmma_gemm<<<nRowTiles, 256, ldsH, stream>>>(x, W0, bufA, CH, CH, CH);
    k_zero<<<zeroBlks, 256, 0, stream>>>(bufB, nElem);
    k_scatter<<<scatBlks, 256, 0, stream>>>(src, dst, dinv, bufA, bufB);
    k_epilogue<<<rowBlks, 128, 0, stream>>>(bufB, bufA, x, dinv, b0, g0, beta0, 0);

    // ---- layer 1: h_in = bufB, hm = bufA, acc/out = bufC, residual bufB ----
    k_wmma_gemm<<<nRowTiles, 256, ldsH, stream>>>(bufB, W1, bufA, CH, CH, CH);
    k_zero<<<zeroBlks, 256, 0, stream>>>(bufC, nElem);
    k_scatter<<<scatBlks, 256, 0, stream>>>(src, dst, dinv, bufA, bufC);
    k_epilogue<<<rowBlks, 128, 0, stream>>>(bufC, bufA, bufB, dinv, b1, g1, beta1, 1);

    // ---- layer 2: h_in = bufC, hm = bufA, acc/out = bufB, residual bufC ----
    k_wmma_gemm<<<nRowTiles, 256, ldsH, stream>>>(bufC, W2, bufA, CH, CH, CH);
    k_zero<<<zeroBlks, 256, 0, stream>>>(bufB, nElem);
    k_scatter<<<scatBlks, 256, 0, stream>>>(src, dst, dinv, bufA, bufB);
    k_epilogue<<<rowBlks, 128, 0, stream>>>(bufB, bufA, bufC, dinv, b2, g2, beta2, 1);

    // ---- output head: logits = bufB @ W_out (128x40), 3 waves (48 padded cols)
    k_wmma_gemm<<<nRowTiles, 96, ldsO, stream>>>(bufB, W_out, bufA, NOUT, NOUT, NOUT);
    k_logsoftmax<<<rowBlks, 128, 0, stream>>>(bufA, b_out, out);
}